// ViT_28415503630629
// MI455X (gfx1250) — compile-verified
//
#include <hip/hip_runtime.h>
#include <hip/hip_bf16.h>
#include <math.h>

typedef __attribute__((ext_vector_type(2))) float v2f;
typedef __attribute__((ext_vector_type(8))) float v8f;

// Workspace layout (floats):
//   [0..63]    u_self      (W_k_self^T folded with qs0)
//   [64..127]  u_aux
//   [128..191] x0 = cls_token + pos_emb[0,0,:]
//   [192..207] aux-branch scalars: m0,m1,mb,S00,S01,S11,S0b,S1b,Sbb,p0,p1,pb,sgu,sbu
//   [208..208+8*2049) logits s[b][k]
#define WS_USEL   0
#define WS_UAUX   64
#define WS_X0     128
#define WS_SCAL   192
#define WS_LOGITS 208

// ---------------------------------------------------------------------------
// Kernel A: batch-independent precompute (cls query path + aux-LN moments)
// ---------------------------------------------------------------------------
__global__ __launch_bounds__(64) void vit_precompute(
    const float* __restrict__ cls_token, const float* __restrict__ cls_aux,
    const float* __restrict__ pos_emb,   const float* __restrict__ ln_g,
    const float* __restrict__ ln_b,      const float* __restrict__ W_qkv,
    const float* __restrict__ W_qkv_aux, const float* __restrict__ W_aux,
    const float* __restrict__ b_aux,     float* __restrict__ ws)
{
    __shared__ float sx0[64], snaf[64], sq[64], sqa[64], sus[64], sua[64];
    __shared__ float r1[64], r2[64];
    const int t = threadIdx.x;

    float x0v = cls_token[t] + pos_emb[t];       // pos_emb[0,0,t]
    sx0[t] = x0v;
    float ca = cls_aux[t];
    r1[t] = ca; r2[t] = ca * ca;
    __syncthreads();
    for (int s = 32; s > 0; s >>= 1) {
        if (t < s) { r1[t] += r1[t + s]; r2[t] += r2[t + s]; }
        __syncthreads();
    }
    float mu   = r1[0] * (1.f / 64.f);
    float ex2  = r2[0] * (1.f / 64.f);
    float rstd = rsqrtf(ex2 - mu * mu + 1e-5f);
    snaf[t] = (ca - mu) * rstd * ln_g[t] + ln_b[t];
    __syncthreads();

    // qs0 / qa0 (query columns 0..63 of the qkv projections)
    float q = 0.f, qa = 0.f;
    for (int i = 0; i < 64; i++) {
        q  += sx0[i]  * W_qkv[i * 192 + t];
        qa += snaf[i] * W_qkv_aux[i * 192 + t];
    }
    sq[t] = q; sqa[t] = qa;
    __syncthreads();

    // u = W_k @ q  (key columns 64..127)
    float us = 0.f, ua = 0.f;
    for (int j = 0; j < 64; j++) {
        us += W_qkv[t * 192 + 64 + j]     * sq[j];
        ua += W_qkv_aux[t * 192 + 64 + j] * sqa[j];
    }
    sus[t] = us; sua[t] = ua;
    ws[WS_USEL + t] = us;
    ws[WS_UAUX + t] = ua;
    ws[WS_X0 + t]   = x0v;
    __syncthreads();

    if (t == 0) {
        float m0=0,m1=0,mb=0,S00=0,S01=0,S11=0,S0b=0,S1b=0,Sbb=0;
        float p0=0,p1=0,pb=0,sgu=0,sbu=0;
        for (int i = 0; i < 64; i++) {
            float a0 = W_aux[i], a1 = W_aux[64 + i], bb = b_aux[i];
            float g = ln_g[i], lb = ln_b[i], u = sua[i];
            m0 += a0; m1 += a1; mb += bb;
            S00 += a0*a0; S01 += a0*a1; S11 += a1*a1;
            S0b += a0*bb; S1b += a1*bb; Sbb += bb*bb;
            p0 += a0*g*u; p1 += a1*g*u; pb += bb*g*u;
            sgu += g*u; sbu += lb*u;
        }
        const float inv64 = 1.f / 64.f;
        float* sc = ws + WS_SCAL;
        sc[0]=m0*inv64;  sc[1]=m1*inv64;  sc[2]=mb*inv64;
        sc[3]=S00*inv64; sc[4]=S01*inv64; sc[5]=S11*inv64;
        sc[6]=S0b*inv64; sc[7]=S1b*inv64; sc[8]=Sbb*inv64;
        sc[9]=p0; sc[10]=p1; sc[11]=pb; sc[12]=sgu; sc[13]=sbu;

        // logit for position 0 (cls attends to cls), same for every batch
        float ds = 0.f, da = 0.f;
        for (int i = 0; i < 64; i++) { ds += sx0[i]*sus[i]; da += snaf[i]*sua[i]; }
        float s0 = ds * da * 0.015625f;            // SCALE^2 = 1/64
        for (int b = 0; b < 8; b++) ws[WS_LOGITS + b * 2049] = s0;
    }
}

// ---------------------------------------------------------------------------
// Kernel B: WMMA logits. One wave per 16-row tile; D = A(16x64) @ B(64x16)
// via 16 chained V_WMMA_F32_16X16X4_F32 steps; B column 0 carries u_self.
// No EXEC-divergent control flow inside the WMMA loop: the B operand is
// zeroed on lanes r!=0 with an arithmetic flag (v_cndmask), not a branch.
// ---------------------------------------------------------------------------
__global__ __launch_bounds__(32) void vit_logits_wmma(
    const float* __restrict__ series, const float* __restrict__ attn_feat,
    const float* __restrict__ pos_emb, float* __restrict__ ws)
{
    const float* usel = ws + WS_USEL;
    const float* sc   = ws + WS_SCAL;
    float* logits     = ws + WS_LOGITS;

    const int blk  = blockIdx.x;
    const int b    = blk >> 7;       // 128 tiles per batch
    const int tile = blk & 127;
    const int lane = threadIdx.x;
    const int r    = lane & 15;      // M = r (both lane halves)
    const int half = lane >> 4;      // half 0: K+{0,1}; half 1: K+{2,3}
    const int k0   = 1 + tile * 16;  // first sequence position in this tile

    const float flag = (r == 0) ? 1.f : 0.f;   // B column 0 only

    const float* arow = series + ((size_t)(b * 2048 + tile * 16 + r)) * 64;
    const float* prow = pos_emb + (size_t)(k0 + r) * 64;

    v8f acc = {0.f,0.f,0.f,0.f,0.f,0.f,0.f,0.f};
    #pragma unroll
    for (int kk = 0; kk < 16; kk++) {
        const int c = kk * 4 + half * 2;        // even -> 8B aligned
        v2f av = *(const v2f*)(arow + c);
        v2f pv = *(const v2f*)(prow + c);
        av.x += pv.x; av.y += pv.y;             // x_k = series + pos
        v2f uv = *(const v2f*)(usel + c);       // broadcast load, same line
        v2f bv; bv.x = uv.x * flag; bv.y = uv.y * flag;
        acc = __builtin_amdgcn_wmma_f32_16x16x4_f32(
            false, av, false, bv, (short)0, acc, false, false);
    }

    // D column N=0: lane 0 holds rows 0..7 (acc[v] -> M=v), lane 16 rows 8..15.
    if (r == 0) {
        const float m0=sc[0], m1=sc[1], mb=sc[2], S00=sc[3], S01=sc[4], S11=sc[5],
                    S0b=sc[6], S1b=sc[7], Sbb=sc[8], p0=sc[9], p1=sc[10], pb=sc[11],
                    sgu=sc[12], sbu=sc[13];
        const int mbase = half * 8;
        #pragma unroll
        for (int v = 0; v < 8; v++) {
            const int k = k0 + mbase + v;
            const size_t fi = ((size_t)(b * 2048 + (k - 1))) * 2;
            const float f0 = attn_feat[fi], f1 = attn_feat[fi + 1];
            // closed-form LayerNorm of af = f0*A0 + f1*A1 + b_aux, dotted with u_aux
            const float mu   = f0*m0 + f1*m1 + mb;
            const float ex2  = f0*f0*S00 + 2.f*f0*f1*S01 + f1*f1*S11
                             + 2.f*f0*S0b + 2.f*f1*S1b + Sbb;
            const float rstd = rsqrtf(ex2 - mu*mu + 1e-5f);
            const float daux = rstd * (f0*p0 + f1*p1 + pb - mu*sgu) + sbu;
            logits[b * 2049 + k] = acc[v] * daux * 0.015625f;   // * SCALE^2
        }
    }
}

// ---------------------------------------------------------------------------
// Kernel C: per-batch softmax over 2049 logits, weighted row-sum, MLP, head.
// ---------------------------------------------------------------------------
__global__ __launch_bounds__(256) void vit_finalize(
    const float* __restrict__ series, const float* __restrict__ pos_emb,
    const float* __restrict__ ws,     const float* __restrict__ W_ff,
    const float* __restrict__ b_ff,   const float* __restrict__ W_head,
    const float* __restrict__ b_head, float* __restrict__ out)
{
    __shared__ float red[256];
    __shared__ float w[2049];
    __shared__ float o0[64];
    __shared__ float h[64];
    const int b = blockIdx.x, t = threadIdx.x;
    const float* x0     = ws + WS_X0;
    const float* logits = ws + WS_LOGITS + b * 2049;

    float mx = -3.4e38f;
    for (int k = t; k < 2049; k += 256) mx = fmaxf(mx, logits[k]);
    red[t] = mx; __syncthreads();
    for (int s = 128; s > 0; s >>= 1) {
        if (t < s) red[t] = fmaxf(red[t], red[t + s]);
        __syncthreads();
    }
    mx = red[0]; __syncthreads();

    float sm = 0.f;
    for (int k = t; k < 2049; k += 256) {
        float e = __expf(logits[k] - mx);
        w[k] = e; sm += e;
    }
    red[t] = sm; __syncthreads();
    for (int s = 128; s > 0; s >>= 1) {
        if (t < s) red[t] += red[t + s];
        __syncthreads();
    }
    const float inv = 1.f / red[0];
    __syncthreads();

    // out0[d] = sum_k w[k] * x[b,k,d]   (256 threads = 4 row-groups x 64 dims)
    const int g = t >> 6, d = t & 63;
    float acc = 0.f;
    for (int k = g; k < 2049; k += 4) {
        float xv = (k == 0) ? x0[d]
                            : (series[((size_t)(b * 2048 + k - 1)) * 64 + d]
                               + pos_emb[(size_t)k * 64 + d]);
        acc += w[k] * xv;
    }
    red[t] = acc; __syncthreads();
    if (t < 64) o0[t] = (red[t] + red[64 + t] + red[128 + t] + red[192 + t]) * inv;
    __syncthreads();

    if (t < 64) {
        float s = b_ff[t];
        for (int i = 0; i < 64; i++) s += o0[i] * W_ff[i * 64 + t];
        h[t] = fmaxf(s, 0.f);
    }
    __syncthreads();
    if (t < 4) {
        float y = b_head[t];
        for (int m = 0; m < 64; m++) y += h[m] * W_head[m * 4 + t];
        out[b * 4 + t] = y;
    }
}

// ---------------------------------------------------------------------------
extern "C" void kernel_launch(void* const* d_in, const int* in_sizes, int n_in,
                              void* d_out, int out_size, void* d_ws, size_t ws_size,
                              hipStream_t stream) {
    const float* series    = (const float*)d_in[0];
    const float* attn_feat = (const float*)d_in[1];
    const float* W_aux     = (const float*)d_in[2];
    const float* b_aux     = (const float*)d_in[3];
    const float* pos_emb   = (const float*)d_in[4];
    const float* cls_token = (const float*)d_in[5];
    const float* cls_aux   = (const float*)d_in[6];
    const float* ln_g      = (const float*)d_in[7];
    const float* ln_b      = (const float*)d_in[8];
    const float* W_qkv     = (const float*)d_in[9];
    const float* W_qkv_aux = (const float*)d_in[10];
    const float* W_ff      = (const float*)d_in[11];
    const float* b_ff      = (const float*)d_in[12];
    const float* W_head    = (const float*)d_in[13];
    const float* b_head    = (const float*)d_in[14];
    float* ws  = (float*)d_ws;
    float* out = (float*)d_out;

    vit_precompute<<<1, 64, 0, stream>>>(cls_token, cls_aux, pos_emb, ln_g, ln_b,
                                         W_qkv, W_qkv_aux, W_aux, b_aux, ws);
    vit_logits_wmma<<<8 * 128, 32, 0, stream>>>(series, attn_feat, pos_emb, ws);
    vit_finalize<<<8, 256, 0, stream>>>(series, pos_emb, ws, W_ff, b_ff,
                                        W_head, b_head, out);
}